// HemorrhageNet_39324720562507
// MI455X (gfx1250) — compile-verified
//
#include <hip/hip_runtime.h>
#include <cstdint>
#include <cstddef>

// Poisson-binomial truncated count distribution -> severity buckets.
// B ~ 1M rows, 96 probabilities per row (x1|x2|x3, 32 each), out (B,5) f32.
//
// Memory-bound (~424 MB total). Strategy: async-DMA (GLOBAL_LOAD_ASYNC_TO_LDS)
// stages 128 rows x 96 f32 per block into LDS with perfectly coalesced b128
// lanes; each wave32 lane then consumes its own row from LDS and runs the
// 96-step recurrence in registers. Output staged via LDS for coalesced stores.

#define ROWS    128   // rows per block == blockDim.x
#define NC      32    // columns per input tensor
#define NSRC    3
#define STRIDE  36    // f32 per row in LDS (144B: 16B aligned, 2-way-max banks)
#define SMEM_FLOATS (NSRC * ROWS * STRIDE + ROWS * 5)

__global__ __launch_bounds__(ROWS)
void HemorrhageNet_sev_kernel(const float* __restrict__ x1,
                              const float* __restrict__ x2,
                              const float* __restrict__ x3,
                              float* __restrict__ out,
                              int nRows)
{
    __shared__ float smem[SMEM_FLOATS];   // single shared object -> LDS offset 0

    const int tid       = threadIdx.x;
    const int blockRow0 = blockIdx.x * ROWS;
    const int rowsHere  = min(ROWS, nRows - blockRow0);

    // ---- Stage 1: async copy 3 tiles of ROWS x 32 f32 into LDS -------------
    // Each tile = 1024 float4; 128 threads x 8 iterations, fully coalesced.
    const float* srcs[NSRC] = { x1, x2, x3 };
    const int maxF4 = rowsHere * (NC / 4) - 1;   // clamp for tail blocks

    #pragma unroll
    for (int t = 0; t < NSRC; ++t) {
        const float* src = srcs[t] + (size_t)blockRow0 * NC;
        #pragma unroll
        for (int i = 0; i < (ROWS * NC / 4) / ROWS; ++i) {   // 8
            int f4 = i * ROWS + tid;          // float4 index within tile
            int r  = f4 >> 3;                 // row (8 float4 per row)
            int c4 = f4 & 7;                  // float4 within row
            unsigned ldsOff = (unsigned)((t * ROWS * STRIDE + r * STRIDE) * 4
                                         + c4 * 16);
            int g4 = (f4 > maxF4) ? maxF4 : f4;
            uint64_t gaddr = (uint64_t)(uintptr_t)(src + (size_t)g4 * 4);
            // CDNA5 async LDS-DMA load, tracked on ASYNCcnt (CDNA5 ISA).
            asm volatile("global_load_async_to_lds_b128 %0, %1, off"
                         :: "v"(ldsOff), "v"(gaddr)
                         : "memory");
        }
    }
    asm volatile("s_wait_asynccnt 0" ::: "memory");
    __syncthreads();

    // ---- Stage 2: per-lane 96-step recurrence on c[0..5] -------------------
    // new_i = c_i + p*(c_{i-1} - c_i); bucket 5 absorbing: new_5 = c_5 + p*c_4
    float c0 = 1.f, c1 = 0.f, c2 = 0.f, c3 = 0.f, c4 = 0.f, c5 = 0.f;

    #pragma unroll
    for (int t = 0; t < NSRC; ++t) {
        const float* row = &smem[(t * ROWS + tid) * STRIDE];
        #pragma unroll
        for (int k = 0; k < NC; ++k) {
            float p  = row[k];
            float n0 = c0 - p * c0;
            float n1 = c1 + p * (c0 - c1);
            float n2 = c2 + p * (c1 - c2);
            float n3 = c3 + p * (c2 - c3);
            float n4 = c4 + p * (c3 - c4);
            float n5 = c5 + p * c4;
            c0 = n0; c1 = n1; c2 = n2; c3 = n3; c4 = n4; c5 = n5;
        }
    }

    // ---- Stage 3: severity mapping, staged through LDS for coalesced out ---
    float* sev = &smem[NSRC * ROWS * STRIDE];
    if (tid < rowsHere) {
        sev[tid * 5 + 0] = c0;
        sev[tid * 5 + 1] = c1 + c2;
        sev[tid * 5 + 2] = c3 + c4;
        sev[tid * 5 + 3] = c5;
        sev[tid * 5 + 4] = 0.f;
    }
    __syncthreads();

    float* outB = out + (size_t)blockRow0 * 5;
    const int lim = rowsHere * 5;
    #pragma unroll
    for (int i = 0; i < 5; ++i) {
        int f = i * ROWS + tid;               // lane-contiguous f32 stores
        if (f < lim) outB[f] = sev[f];
    }
}

extern "C" void kernel_launch(void* const* d_in, const int* in_sizes, int n_in,
                              void* d_out, int out_size, void* d_ws, size_t ws_size,
                              hipStream_t stream) {
    (void)n_in; (void)out_size; (void)d_ws; (void)ws_size;
    const float* x1 = (const float*)d_in[0];
    const float* x2 = (const float*)d_in[1];
    const float* x3 = (const float*)d_in[2];
    float* out = (float*)d_out;

    const int nRows = in_sizes[0] / NC;                 // B
    const int grid  = (nRows + ROWS - 1) / ROWS;        // 8192 for B=1M
    HemorrhageNet_sev_kernel<<<grid, ROWS, 0, stream>>>(x1, x2, x3, out, nRows);
}